// Net_RGCN_41927470743854
// MI455X (gfx1250) — compile-verified
//
#include <hip/hip_runtime.h>
#include <hip/hip_bf16.h>
#include <math.h>

// Problem constants from the reference
#define RGCN_OUT   300
#define RGCN_R     5
#define RGCN_B     5

typedef __attribute__((ext_vector_type(2))) float v2f;
typedef __attribute__((ext_vector_type(8))) float v8f;

// ---------------------------------------------------------------------------
// Zero-fill workspace (agg + cnt). Kernel instead of memset: graph-capture safe.
// ---------------------------------------------------------------------------
__global__ void rgcn_zero(float* __restrict__ p, long long n) {
    long long i = (long long)blockIdx.x * blockDim.x + threadIdx.x;
    long long stride = (long long)gridDim.x * blockDim.x;
    for (; i < n; i += stride) p[i] = 0.0f;
}

// ---------------------------------------------------------------------------
// Stage 1: weight[r, p] = sum_b comp[r,b] * basis[b, p],  p in [0, NP)
// NP = N_NODES*OUT (flattened). Mapped onto V_WMMA_F32_16X16X4_F32:
//   D(16x16) = A(16x4) x B(4x16) + C
//   M = r (rows 0..4 live, 5..15 zero), N = 16 consecutive p, K = bases.
//   K=5 -> two chained WMMAs: bases {0..3} then base {4} (zero-padded chunk).
// A layout (32-bit A 16x4): lane m=lane&15; lanes<16 hold K=(0,1) in (x,y),
// lanes>=16 hold K=(2,3). B mirrors with N=lane&15.
// D layout: VGPR k holds M=k (lanes 0-15) / M=k+8 (lanes 16-31), N=lane&15.
//
// Memory-bound (300MB in / 300MB out, ~26us each way at 23.3 TB/s), so WMMA is
// free throughput-wise. Main loop is guard-free: chunk-1 B row is loaded from
// row b=4 for ALL lanes (in-bounds; lanes>=16 see A==0 and 0*finite==0, so the
// bogus row contributes nothing). Basis loads are non-temporal: each position
// is read exactly once, and we want L2 to keep weight lines for the gather.
// ---------------------------------------------------------------------------
__global__ void rgcn_einsum_wmma(const float* __restrict__ basis,
                                 const float* __restrict__ comp,
                                 float* __restrict__ weight,
                                 long long NP) {
    const int lane = threadIdx.x & 31;
    const int wid  = ((int)(blockIdx.x * blockDim.x + threadIdx.x)) >> 5;
    const int nwaves = (int)((gridDim.x * blockDim.x) >> 5);
    const long long ntiles = NP >> 4;              // full 16-wide tiles

    const int  m  = lane & 15;
    const bool lo = lane < 16;
    const int  k0 = lo ? 0 : 2;
    const int  k1 = lo ? 1 : 3;

    // A = comp, constant across tiles (zero-pad rows r>=5, bases b>=5)
    v2f a0, a1;
    a0.x = (m < RGCN_R) ? comp[m * RGCN_B + k0] : 0.0f;
    a0.y = (m < RGCN_R) ? comp[m * RGCN_B + k1] : 0.0f;
    a1.x = (m < RGCN_R && lo) ? comp[m * RGCN_B + 4] : 0.0f;  // chunk1: b=4 only
    a1.y = 0.0f;

    const float* row0 = basis + (long long)k0 * NP;
    const float* row1 = basis + (long long)k1 * NP;
    const float* row4 = basis + 4ll * NP;

    for (long long t = wid; t < ntiles; t += nwaves) {
        const long long p = (t << 4) + m;
        v2f b0, b1;
        b0.x = __builtin_nontemporal_load(row0 + p);
        b0.y = __builtin_nontemporal_load(row1 + p);
        b1.x = __builtin_nontemporal_load(row4 + p);   // lanes>=16: killed by A==0
        b1.y = 0.0f;

        v8f c = {};
#if __has_builtin(__builtin_amdgcn_wmma_f32_16x16x4_f32)
        c = __builtin_amdgcn_wmma_f32_16x16x4_f32(false, a0, false, b0,
                                                  (short)0, c, false, false);
        c = __builtin_amdgcn_wmma_f32_16x16x4_f32(false, a1, false, b1,
                                                  (short)0, c, false, false);
#else
        // Scalar fallback (host parse / builtin unavailable).
        if (lo) {
            #pragma unroll
            for (int r = 0; r < 8; ++r) {
                float acc = 0.0f;
                if (r < RGCN_R) {
                    #pragma unroll
                    for (int b = 0; b < RGCN_B; ++b)
                        acc += comp[r * RGCN_B + b] * basis[(long long)b * NP + p];
                }
                c[r] = acc;
            }
        }
#endif
        if (lo) {
            #pragma unroll
            for (int r = 0; r < RGCN_R; ++r)        // D rows M=0..4 live in lanes 0-15
                weight[(long long)r * NP + p] = c[r];
        }
    }

    // Scalar tail for NP % 16 != 0 (never taken at reference sizes).
    if (wid == 0) {
        const long long base = ntiles << 4;
        const long long p = base + lane;
        if (p < NP) {
            #pragma unroll
            for (int r = 0; r < RGCN_R; ++r) {
                float acc = 0.0f;
                #pragma unroll
                for (int b = 0; b < RGCN_B; ++b)
                    acc += comp[r * RGCN_B + b] * basis[(long long)b * NP + p];
                weight[(long long)r * NP + p] = acc;
            }
        }
    }
}

// ---------------------------------------------------------------------------
// Stage 2a (materialized path): wave per edge, gather weight row, atomic scatter
// ---------------------------------------------------------------------------
__global__ void rgcn_scatter_mat(const int* __restrict__ src,
                                 const int* __restrict__ dst,
                                 const int* __restrict__ etype,
                                 const float* __restrict__ weight,
                                 float* __restrict__ agg,
                                 float* __restrict__ cnt,
                                 int E, long long NP) {
    const int gid  = blockIdx.x * blockDim.x + threadIdx.x;
    const int e    = gid >> 5;
    const int lane = gid & 31;
    if (e >= E) return;
    const int s = src[e], d = dst[e], t = etype[e];
    const float* wrow = weight + (long long)t * NP + (long long)s * RGCN_OUT;
    float*       arow = agg + (long long)d * RGCN_OUT;
    __builtin_prefetch(wrow + lane * 4, 0, 0);          // global_prefetch_b8
    if (lane == 0) atomicAdd(&cnt[d], 1.0f);
    for (int k = lane; k < RGCN_OUT; k += 32)
        atomicAdd(&arow[k], wrow[k]);
}

// ---------------------------------------------------------------------------
// Stage 2b (fused fallback if workspace too small for the 300MB weight):
// message computed on the fly from basis + comp[edge_type]
// ---------------------------------------------------------------------------
__global__ void rgcn_scatter_fused(const int* __restrict__ src,
                                   const int* __restrict__ dst,
                                   const int* __restrict__ etype,
                                   const float* __restrict__ basis,
                                   const float* __restrict__ comp,
                                   float* __restrict__ agg,
                                   float* __restrict__ cnt,
                                   int E, long long NP) {
    const int gid  = blockIdx.x * blockDim.x + threadIdx.x;
    const int e    = gid >> 5;
    const int lane = gid & 31;
    if (e >= E) return;
    const int s = src[e], d = dst[e], t = etype[e];
    const float c0 = comp[t * RGCN_B + 0], c1 = comp[t * RGCN_B + 1];
    const float c2 = comp[t * RGCN_B + 2], c3 = comp[t * RGCN_B + 3];
    const float c4 = comp[t * RGCN_B + 4];
    const float* brow = basis + (long long)s * RGCN_OUT;
    float*       arow = agg + (long long)d * RGCN_OUT;
    if (lane == 0) atomicAdd(&cnt[d], 1.0f);
    for (int k = lane; k < RGCN_OUT; k += 32) {
        float msg = c0 * brow[k]
                  + c1 * brow[NP + k]
                  + c2 * brow[2 * NP + k]
                  + c3 * brow[3 * NP + k]
                  + c4 * brow[4 * NP + k];
        atomicAdd(&arow[k], msg);
    }
}

// ---------------------------------------------------------------------------
// Stage 3: wave per node: mean, +root+bias, relu, log_softmax over 300 dims.
// wave32 shuffle reductions (offsets 16..1).
// ---------------------------------------------------------------------------
__global__ void rgcn_finalize(const float* __restrict__ agg,
                              const float* __restrict__ cnt,
                              const float* __restrict__ root,
                              const float* __restrict__ bias,
                              float* __restrict__ out, int Nn) {
    const int gid  = blockIdx.x * blockDim.x + threadIdx.x;
    const int n    = gid >> 5;
    const int lane = gid & 31;
    if (n >= Nn) return;
    const float inv = 1.0f / fmaxf(cnt[n], 1.0f);
    const float* arow = agg  + (long long)n * RGCN_OUT;
    const float* rrow = root + (long long)n * RGCN_OUT;

    float h[(RGCN_OUT + 31) / 32];
    int   nk = 0;
    float mx = -1e30f;
    for (int k = lane, i = 0; k < RGCN_OUT; k += 32, ++i) {
        float v = fmaxf(arow[k] * inv + rrow[k] + bias[k], 0.0f);  // relu
        h[i] = v;
        mx = fmaxf(mx, v);
        nk = i + 1;
    }
    #pragma unroll
    for (int off = 16; off > 0; off >>= 1)
        mx = fmaxf(mx, __shfl_xor(mx, off, 32));

    float se = 0.0f;
    for (int i = 0; i < nk; ++i) se += expf(h[i] - mx);
    #pragma unroll
    for (int off = 16; off > 0; off >>= 1)
        se += __shfl_xor(se, off, 32);
    const float ls = logf(se);

    for (int k = lane, i = 0; k < RGCN_OUT; k += 32, ++i)
        out[(long long)n * RGCN_OUT + k] = h[i] - mx - ls;
}

// ---------------------------------------------------------------------------
extern "C" void kernel_launch(void* const* d_in, const int* in_sizes, int n_in,
                              void* d_out, int out_size, void* d_ws, size_t ws_size,
                              hipStream_t stream) {
    const int*   edge_index = (const int*)d_in[0];     // [2, E]
    const int*   edge_type  = (const int*)d_in[1];     // [E]
    /* d_in[2] = edge_norm: unused by the reference */
    const float* basis      = (const float*)d_in[3];   // [B, N, OUT]
    const float* comp       = (const float*)d_in[4];   // [R, B]
    const float* root       = (const float*)d_in[5];   // [N, OUT]
    const float* bias       = (const float*)d_in[6];   // [OUT]
    float*       out        = (float*)d_out;

    const int E  = in_sizes[1];
    const int Nn = in_sizes[5] / RGCN_OUT;
    const long long NP = (long long)Nn * RGCN_OUT;     // flattened (n, d)

    const int* src = edge_index;
    const int* dst = edge_index + E;

    float* agg    = (float*)d_ws;                      // [N, OUT]
    float* cnt    = agg + NP;                          // [N]
    float* weight = cnt + Nn;                          // [R, N, OUT] if it fits
    const size_t need_full =
        (size_t)(NP + Nn + (long long)RGCN_R * NP) * sizeof(float);
    const bool materialize = (ws_size >= need_full);

    // zero agg + cnt
    rgcn_zero<<<8192, 256, 0, stream>>>(agg, NP + Nn);

    if (materialize) {
        // Stage 1: WMMA basis contraction -> weight (600MB streamed, mem-bound)
        rgcn_einsum_wmma<<<4096, 256, 0, stream>>>(basis, comp, weight, NP);
        // Stage 2: gather + atomic scatter (wave per edge, 8 edges/block)
        const int blocks_e = (E + 7) / 8;
        rgcn_scatter_mat<<<blocks_e, 256, 0, stream>>>(src, dst, edge_type,
                                                       weight, agg, cnt, E, NP);
    } else {
        const int blocks_e = (E + 7) / 8;
        rgcn_scatter_fused<<<blocks_e, 256, 0, stream>>>(src, dst, edge_type,
                                                         basis, comp, agg, cnt,
                                                         E, NP);
    }

    // Stage 3: mean + root + bias + relu + log_softmax (wave per node)
    const int blocks_n = (Nn + 7) / 8;
    rgcn_finalize<<<blocks_n, 256, 0, stream>>>(agg, cnt, root, bias, out, Nn);
}